// random_conv_19731079757980
// MI455X (gfx1250) — compile-verified
//
#include <hip/hip_runtime.h>
#include <math.h>

typedef __attribute__((ext_vector_type(2))) float v2f;
typedef __attribute__((ext_vector_type(8))) float v8f;

#define IMG_H 128
#define IMG_W 128
#define ROWS_PER_BLOCK 8
#define TILE_ROWS (ROWS_PER_BLOCK + 2)   // 10 rows incl. halo
#define TILE_COLS 130                    // 128 + halo
#define LDS_PITCH 132                    // padded pitch

#if __has_builtin(__builtin_amdgcn_global_load_async_to_lds_b32)
#define HAVE_ASYNC_LDS 1
#else
#define HAVE_ASYNC_LDS 0
#endif

#if HAVE_ASYNC_LDS
// Generic->AS casts via integers: AS1 (global) is 64-bit; AS3 (LDS) is 32-bit,
// and the low 32 bits of a generic LDS address are the LDS byte offset.
__device__ __forceinline__ __attribute__((address_space(1))) int*
to_global(const void* p) {
    return (__attribute__((address_space(1))) int*)(uintptr_t)p;
}
__device__ __forceinline__ __attribute__((address_space(3))) int*
to_local(void* p) {
    return (__attribute__((address_space(3))) int*)(uint32_t)(uintptr_t)p;
}
#endif

// Tap k (0..26) -> LDS element offset of (channel k/9, row (k%9)/3, col k%3),
// relative to a lane's (row, col) base. Pure compile-time.
__device__ constexpr int tap_off(int k) {
    return ((k / 9) * TILE_ROWS + ((k % 9) / 3)) * LDS_PITCH + (k % 3);
}

// One block = 8 waves = one (n, frame, 8-row slab). Each wave: one output row.
// Implicit GEMM per 16-pixel strip with V_WMMA_F32_16X16X4_F32:
//   A(16x4) = weights/255 (M=out-channel, K=tap), B(4x16) = raw patches
//   (N=pixel), 7 K-steps of 4 (27 taps padded to 28).
__global__ __launch_bounds__(256) void conv3x3_sigmoid_wmma(
    const float* __restrict__ x,     // (N, 9, 128, 128)
    const float* __restrict__ wgt,   // (N, 3, 3, 3, 3) OIHW per frame-triple
    float* __restrict__ out)         // (N, 9, 128, 128)
{
    __shared__ float lds[3 * TILE_ROWS * LDS_PITCH];

    const int h0 = blockIdx.x * ROWS_PER_BLOCK;   // first output row of slab
    const int f  = blockIdx.y;                    // frame 0..2
    const int n  = blockIdx.z;                    // sample

    const float* __restrict__ xin = x   + ((size_t)n * 9 + (size_t)f * 3) * (IMG_H * IMG_W);
    const float* __restrict__ wn  = wgt + (size_t)n * 81;
    float* __restrict__ obase     = out + ((size_t)n * 9 + (size_t)f * 3) * (IMG_H * IMG_W);

    // ---- stage raw input: 3ch x 10rows x 130cols, edge-clamped ----
    const int total = 3 * TILE_ROWS * TILE_COLS;
    for (int idx = threadIdx.x; idx < total; idx += 256) {
        int c  = idx % TILE_COLS;
        int t  = idx / TILE_COLS;
        int r  = t % TILE_ROWS;
        int ch = t / TILE_ROWS;
        int gr = h0 - 1 + r; gr = gr < 0 ? 0 : (gr > IMG_H - 1 ? IMG_H - 1 : gr);
        int gc = c - 1;      gc = gc < 0 ? 0 : (gc > IMG_W - 1 ? IMG_W - 1 : gc);
        const float* gsrc = xin + ch * (IMG_H * IMG_W) + gr * IMG_W + gc;
        float*       ldst = &lds[(ch * TILE_ROWS + r) * LDS_PITCH + c];
#if HAVE_ASYNC_LDS
        __builtin_amdgcn_global_load_async_to_lds_b32(to_global(gsrc), to_local(ldst), 0, 0);
#else
        *ldst = *gsrc;
#endif
    }
#if HAVE_ASYNC_LDS
#if __has_builtin(__builtin_amdgcn_s_wait_asynccnt)
    __builtin_amdgcn_s_wait_asynccnt(0);
#else
    asm volatile("s_wait_asynccnt 0x0" ::: "memory");
#endif
#endif
    __syncthreads();

    const int lane = threadIdx.x & 31;
    const int wr   = threadIdx.x >> 5;    // wave id = row within slab
    const int m16  = lane & 15;           // A: M row (channel) / B: N col (pixel)
    const int half = lane >> 4;           // K sub-offset selector (0 or 1)

    // ---- per-lane setup: weights (A) and 14 pinned LDS tap offsets (B) ----
    const int   lbase  = wr * LDS_PITCH + m16;            // lane row/col base
    const int   chc    = (m16 < 3) ? m16 : 0;             // clamp: no OOB reads
    const float wscale = (m16 < 3) ? (1.0f / 255.0f) : 0.0f;
    const float* __restrict__ wb = wn + chc * 27;

    float wA[7][2];
    int   off[7][2];
#pragma unroll
    for (int ks = 0; ks < 7; ++ks) {
#pragma unroll
        for (int j = 0; j < 2; ++j) {
            const int k0 = ks * 4 + j;                    // tap if half==0 (compile-time)
            const int k1 = (k0 + 2 <= 26) ? k0 + 2 : 26;  // tap if half==1 (clamped pad)
            // B tap offset: select between two compile-time constants.
            int o = lbase + (half ? tap_off(k1) : tap_off(k0));
            asm("" : "+v"(o));                            // pin: forbid rematerialization
            off[ks][j] = o;
            // A value: weight/255; pad tap (k==27) is zero.
            int kidx = (k0 + 2 <= 26) ? (k0 + half * 2) : (half ? 26 : k0);
            float wv = wb[kidx] * wscale;
            if (k0 + 2 > 26) wv = half ? 0.0f : wv;       // compile-time guard
            wA[ks][j] = wv;
        }
    }

    float* __restrict__ orow = obase + (h0 + wr) * IMG_W + m16;

#pragma unroll
    for (int t = 0; t < IMG_W / 16; ++t) {
        v8f acc = {};
#pragma unroll
        for (int ks = 0; ks < 7; ++ks) {
            v2f a; a.x = wA[ks][0];            a.y = wA[ks][1];
            v2f b; b.x = lds[off[ks][0] + t * 16]; b.y = lds[off[ks][1] + t * 16];
            // D = A(16x4, f32) x B(4x16, f32) + C(16x16, f32)
            acc = __builtin_amdgcn_wmma_f32_16x16x4_f32(
                false, a, false, b, (short)0, acc, false, false);
        }
        // D layout: acc[v], lane L -> M = v + 8*(L/16), N = L%16.
        // Channels 0..2 live in acc[0..2], lanes 0..15.
        if (half == 0) {
#pragma unroll
            for (int v = 0; v < 3; ++v) {
                // sigmoid*255 via v_exp + v_rcp (no f32 div sequence)
                float e = __expf(-acc[v]);
                orow[t * 16 + v * (IMG_H * IMG_W)] = 255.0f * __builtin_amdgcn_rcpf(1.0f + e);
            }
        }
    }
}

extern "C" void kernel_launch(void* const* d_in, const int* in_sizes, int n_in,
                              void* d_out, int out_size, void* d_ws, size_t ws_size,
                              hipStream_t stream) {
    const float* x = (const float*)d_in[0];   // (N, 9, 128, 128) f32
    const float* w = (const float*)d_in[1];   // (N, 3, 3, 3, 3)  f32
    float* out = (float*)d_out;

    const int n = in_sizes[1] / 81;           // samples (256)
    dim3 grid(IMG_H / ROWS_PER_BLOCK, 3, n);  // (row-slab, frame, sample)
    conv3x3_sigmoid_wmma<<<grid, 256, 0, stream>>>(x, w, out);
}